// SAGE_PyG_81243601371388
// MI455X (gfx1250) — compile-verified
//
#include <hip/hip_runtime.h>
#include <hip/hip_bf16.h>

typedef float v2f __attribute__((ext_vector_type(2)));
typedef float v8f __attribute__((ext_vector_type(8)));

#define IN_CH 128

// ---------------------------------------------------------------------------
// Zero-fill (float4 grid-stride)
// ---------------------------------------------------------------------------
__global__ void zero_f4(float4* __restrict__ p, long n4) {
    long i = (long)blockIdx.x * blockDim.x + threadIdx.x;
    long stride = (long)gridDim.x * blockDim.x;
    float4 z; z.x = 0.f; z.y = 0.f; z.z = 0.f; z.w = 0.f;
    for (; i < n4; i += stride) p[i] = z;
}

// ---------------------------------------------------------------------------
// Y[rows x DOUT] = X[rows x 128] @ W[128 x DOUT]   using V_WMMA_F32_16X16X4_F32
//
// Block: 32 rows, DOUT cols. One wave per 16-col tile (WAVES = DOUT/16).
// A tile staged in LDS; each wave preloads its 128x16 panel of W into 32 v2f
// B-fragments held in registers, reused across the 2 row tiles.
//
// Fragment layouts per CDNA5 ISA 7.12.2 (wave32):
//   A 16x4 f32 : lane l -> m = l&15, koff = 2*(l>>4); v0 = A[m][k0+koff],
//                v1 = A[m][k0+koff+1]
//   B 4x16 f32 : lane l -> n = l&15, same K split
//   C/D 16x16  : lane l, vgpr j -> D[8*(l>>4)+j][l&15]
// ---------------------------------------------------------------------------
template <int DOUT>
__global__ __launch_bounds__(32 * (DOUT / 16))
void gcn_gemm(const float* __restrict__ X, const float* __restrict__ W,
              float* __restrict__ Y) {
    constexpr int K     = IN_CH;        // 128
    constexpr int WAVES = DOUT / 16;    // 8 or 4
    constexpr int ROWS  = 32;           // rows per block (100000 % 32 == 0)
    constexpr int LDA   = 132;          // LDS row stride (floats): 16B-aligned, spreads banks

    __shared__ float As[ROWS * LDA];

    const int tid  = threadIdx.x;
    const int wave = tid >> 5;
    const int lane = tid & 31;
    const long row0 = (long)blockIdx.x * ROWS;

    // Cooperative load of X tile [ROWS x K] into LDS via float4.
    for (int f = tid; f < ROWS * (K / 4); f += WAVES * 32) {
        const int r  = f >> 5;          // K/4 == 32 float4 per row
        const int c4 = f & 31;
        const float4 v =
            reinterpret_cast<const float4*>(X + (row0 + r) * (long)K)[c4];
        *reinterpret_cast<float4*>(&As[r * LDA + c4 * 4]) = v;
    }
    __syncthreads();

    const int nm   = lane & 15;         // N index (B/D) == M index (A) for this lane
    const int half = lane >> 4;
    const int koff = half * 2;
    const int n0   = wave * 16;

    // Preload B fragments: full 128 x 16 panel of W for this wave's column tile.
    v2f Bf[K / 4];
#pragma unroll
    for (int kc = 0; kc < K / 4; ++kc) {
        const int kk = kc * 4 + koff;
        v2f b;
        b.x = W[(long)kk * DOUT + n0 + nm];
        b.y = W[(long)(kk + 1) * DOUT + n0 + nm];
        Bf[kc] = b;
    }

#pragma unroll
    for (int rt = 0; rt < ROWS / 16; ++rt) {
        v8f acc = {};
        const float* a_base = &As[(rt * 16 + nm) * LDA];
#pragma unroll
        for (int kc = 0; kc < K / 4; ++kc) {
            const int kk = kc * 4 + koff;           // even -> 8B aligned in LDS
            const v2f a = *reinterpret_cast<const v2f*>(&a_base[kk]);
            acc = __builtin_amdgcn_wmma_f32_16x16x4_f32(
                /*neg_a=*/false, a, /*neg_b=*/false, Bf[kc],
                /*c_mod=*/(short)0, acc, /*reuse_a=*/false, /*reuse_b=*/false);
        }
        const int mb = half * 8;
#pragma unroll
        for (int j = 0; j < 8; ++j) {
            Y[(row0 + rt * 16 + mb + j) * (long)DOUT + n0 + nm] = acc[j];
        }
    }
}

// ---------------------------------------------------------------------------
// out[dst[e]] += H[src[e]]  over all edges. One wave per edge; lane covers
// D/32 contiguous floats (vector load from h[src], f32 atomics into out[dst]).
// ---------------------------------------------------------------------------
template <int D>
__global__ __launch_bounds__(256)
void scatter_add(const float* __restrict__ H, const int* __restrict__ src,
                 const int* __restrict__ dst, float* __restrict__ out, int nE) {
    constexpr int VPT = D / 32;   // 4 (D=128) or 2 (D=64)
    const int e = blockIdx.x * (blockDim.x >> 5) + (threadIdx.x >> 5);
    if (e >= nE) return;
    const int lane = threadIdx.x & 31;
    const int s = src[e];
    const int d = dst[e];
    const float* hp = H + (long)s * D + lane * VPT;
    float* op = out + (long)d * D + lane * VPT;
    if constexpr (VPT == 4) {
        const float4 v = *reinterpret_cast<const float4*>(hp);
        atomicAdd(op + 0, v.x);
        atomicAdd(op + 1, v.y);
        atomicAdd(op + 2, v.z);
        atomicAdd(op + 3, v.w);
    } else {
        const float2 v = *reinterpret_cast<const float2*>(hp);
        atomicAdd(op + 0, v.x);
        atomicAdd(op + 1, v.y);
    }
}

// ---------------------------------------------------------------------------
// Launch: h1 = agg(x@W1); h2 = agg(h1@W2); out = agg(h2@W3)
// ws layout: bufA = N*128 floats (GEMM outputs), bufB = N*128 floats (agg).
// ---------------------------------------------------------------------------
extern "C" void kernel_launch(void* const* d_in, const int* in_sizes, int n_in,
                              void* d_out, int out_size, void* d_ws, size_t ws_size,
                              hipStream_t stream) {
    const float* x  = (const float*)d_in[0];
    const int*   ei = (const int*)d_in[1];
    const float* W1 = (const float*)d_in[2];
    const float* W2 = (const float*)d_in[3];
    const float* W3 = (const float*)d_in[4];

    const int nNodes = in_sizes[0] / IN_CH;      // 100000
    const int nE     = in_sizes[1] / 2;          // 1600000
    const int* src = ei;
    const int* dst = ei + nE;

    float* bufA = (float*)d_ws;
    float* bufB = bufA + (size_t)nNodes * 128;
    float* outF = (float*)d_out;

    const int gemmGrid = nNodes / 32;            // 3125
    const int edgeGrid = (nE + 7) / 8;           // 8 edges per 256-thread block
    const dim3 zgrid(4096), zblk(256);

    // ---- Layer 1: t1 = x @ W1 ; h1 = scatter(t1) ----
    zero_f4<<<zgrid, zblk, 0, stream>>>((float4*)bufB, (long)nNodes * 128 / 4);
    gcn_gemm<128><<<gemmGrid, 256, 0, stream>>>(x, W1, bufA);
    scatter_add<128><<<edgeGrid, 256, 0, stream>>>(bufA, src, dst, bufB, nE);

    // ---- Layer 2: t2 = h1 @ W2 ; h2 = scatter(t2) ----
    gcn_gemm<128><<<gemmGrid, 256, 0, stream>>>(bufB, W2, bufA);
    zero_f4<<<zgrid, zblk, 0, stream>>>((float4*)bufB, (long)nNodes * 128 / 4);
    scatter_add<128><<<edgeGrid, 256, 0, stream>>>(bufA, src, dst, bufB, nE);

    // ---- Layer 3: t3 = h2 @ W3 ; out = scatter(t3) ----
    gcn_gemm<64><<<gemmGrid, 128, 0, stream>>>(bufB, W3, bufA);
    zero_f4<<<zgrid, zblk, 0, stream>>>((float4*)outF, (long)nNodes * 64 / 4);
    scatter_add<64><<<edgeGrid, 256, 0, stream>>>(bufA, src, dst, outF, nE);
}